// ResidualVectorQuantizer_80822694576506
// MI455X (gfx1250) — compile-verified
//
#include <hip/hip_runtime.h>

typedef __attribute__((ext_vector_type(2))) float v2f;
typedef __attribute__((ext_vector_type(8))) float v8f;

#define N_CB   8
#define KENT   1024
#define CD     64
#define BZ     8
#define TZ     2048
#define DZ     (N_CB * CD)          // 512
#define TOK_TOTAL (BZ * TZ)         // 16384
#define WAVES  8
#define TOK_PER_WAVE 16
#define TOK_PER_BLOCK (WAVES * TOK_PER_WAVE)   // 128

// ---------------------------------------------------------------------------
// Pass 1: ||cb[e]||^2 for all 8*1024 codebook entries -> d_ws
// ---------------------------------------------------------------------------
__global__ void rvq_cbnorm_kernel(const float* __restrict__ cbs,
                                  float* __restrict__ norms) {
    int e = blockIdx.x * blockDim.x + threadIdx.x;
    if (e < N_CB * KENT) {
        const float* p = cbs + (size_t)e * CD;
        float s = 0.f;
        #pragma unroll
        for (int k = 0; k < CD; ++k) s += p[k] * p[k];
        norms[e] = s;
    }
}

// ---------------------------------------------------------------------------
// Pass 2: the RVQ. One block = 128 tokens (16 per wave). Stages sequential.
// dist tile per wave/stage: 16 tokens x 1024 entries via f32 WMMA 16x16x4.
// ---------------------------------------------------------------------------
__global__ __launch_bounds__(256)
void rvq_main_kernel(const float* __restrict__ z,
                     const float* __restrict__ cbs,
                     const float* __restrict__ cbnorm,
                     float* __restrict__ zq_out,
                     float* __restrict__ idx_out,
                     float* __restrict__ logits_out) {
    __shared__ float resLDS[TOK_PER_BLOCK * CD];   // 32 KB residual tile
    __shared__ float resnorm[TOK_PER_BLOCK];
    __shared__ int   idxLDS[TOK_PER_BLOCK];

    const int tid   = threadIdx.x;
    const int wv    = tid >> 5;
    const int lane  = tid & 31;
    const int hh    = lane >> 4;        // lane half: 0 = lanes 0-15, 1 = 16-31
    const int l16   = lane & 15;
    const int tok0  = blockIdx.x * TOK_PER_BLOCK;  // global token base of block
    const int wtok0 = wv * TOK_PER_WAVE;           // wave's token base in LDS

    // initial residual = z chunk 0 (32 elements per lane, coalesced)
    #pragma unroll
    for (int j = 0; j < TOK_PER_WAVE * CD / 32; ++j) {
        int f  = j * 32 + lane;
        int tl = f >> 6;
        int k  = f & 63;
        resLDS[(wtok0 + tl) * CD + k] =
            z[(size_t)(tok0 + wtok0 + tl) * DZ + k];
    }

    for (int i = 0; i < N_CB; ++i) {
        const float* cb = cbs + (size_t)i * KENT * CD;

        // ||res||^2 per token (2 lanes per token, 32 elems each, shfl-combine)
        {
            float s = 0.f;
            const float* rp = &resLDS[(wtok0 + l16) * CD + hh * 32];
            #pragma unroll
            for (int k = 0; k < 32; ++k) s += rp[k] * rp[k];
            s += __shfl_xor(s, 16, 32);
            if (hh == 0) resnorm[wtok0 + l16] = s;
        }

        float best[8];
        int   bestIdx[8];
        #pragma unroll
        for (int r = 0; r < 8; ++r) { best[r] = 3.4e38f; bestIdx[r] = 0; }

        // 64 N-tiles of 16 entries; K=64 in 16 steps of 4 (f32 WMMA)
        for (int nt = 0; nt < KENT / 16; ++nt) {
            const int n0 = nt * 16;
            const int e  = n0 + l16;
            const float cn = cbnorm[i * KENT + e];
            v8f acc = {};
            const float* arow = &resLDS[(wtok0 + l16) * CD];
            const float* brow = cb + (size_t)e * CD;
            #pragma unroll
            for (int kk = 0; kk < CD / 4; ++kk) {
                const int k = kk * 4 + hh * 2;
                // A 16x4 f32: lanes0-15 K=0/1, lanes16-31 K=2/3 (ISA 7.12.2)
                v2f a; a[0] = arow[k]; a[1] = arow[k + 1];
                // B 4x16 f32: B[k][n] = cb[n][k], same K striping over halves
                v2f b; b[0] = brow[k]; b[1] = brow[k + 1];
                acc = __builtin_amdgcn_wmma_f32_16x16x4_f32(
                    false, a, false, b, (short)0, acc, false, false);
            }
            // C layout: VGPR r -> M=r (lanes0-15) / M=r+8 (lanes16-31), N=l16
            const int mbase = hh * 8;
            #pragma unroll
            for (int r = 0; r < 8; ++r) {
                const int m    = mbase + r;
                const int gtok = tok0 + wtok0 + m;
                const float dist = resnorm[wtok0 + m] + cn - 2.0f * acc[r];
                const int b_ = gtok >> 11;     // / 2048
                const int t_ = gtok & 2047;
                size_t lidx = (((size_t)b_ * N_CB + i) * TZ + t_) * KENT + e;
                __builtin_nontemporal_store(-dist, &logits_out[lidx]);
                if (dist < best[r]) { best[r] = dist; bestIdx[r] = e; }
            }
        }

        // argmin across the 16 lanes of each half (tie-break: lowest index)
        #pragma unroll
        for (int r = 0; r < 8; ++r) {
            #pragma unroll
            for (int off = 8; off >= 1; off >>= 1) {
                float ov = __shfl_xor(best[r], off, 32);
                int   oi = __shfl_xor(bestIdx[r], off, 32);
                if (ov < best[r] || (ov == best[r] && oi < bestIdx[r])) {
                    best[r] = ov; bestIdx[r] = oi;
                }
            }
        }
        if (l16 == 0) {
            #pragma unroll
            for (int r = 0; r < 8; ++r) idxLDS[wtok0 + hh * 8 + r] = bestIdx[r];
        }

        // indices output (stored as float in the concatenated f32 out buffer)
        if (hh == 0) {
            const int gtok = tok0 + wtok0 + l16;
            const int b_ = gtok >> 11;
            const int t_ = gtok & 2047;
            idx_out[((size_t)b_ * N_CB + i) * TZ + t_] =
                (float)idxLDS[wtok0 + l16];
        }

        // gather q, write z_q chunk, update residual for next stage
        #pragma unroll
        for (int j = 0; j < TOK_PER_WAVE * CD / 32; ++j) {
            int f   = j * 32 + lane;
            int tl  = f >> 6;
            int k   = f & 63;
            int tok = wtok0 + tl;
            int eq  = idxLDS[tok];
            float q = cb[(size_t)eq * CD + k];
            int gtok = tok0 + tok;
            zq_out[(size_t)gtok * DZ + i * CD + k] = q;
            if (i < N_CB - 1) {
                resLDS[tok * CD + k] =
                    z[(size_t)gtok * DZ + (i + 1) * CD + k]
                    + resLDS[tok * CD + k] - q;
            }
        }
    }
}

// ---------------------------------------------------------------------------
extern "C" void kernel_launch(void* const* d_in, const int* in_sizes, int n_in,
                              void* d_out, int out_size, void* d_ws, size_t ws_size,
                              hipStream_t stream) {
    const float* z   = (const float*)d_in[0];
    const float* cbs = (const float*)d_in[1];

    float* out  = (float*)d_out;
    float* zq   = out;                                      // B*T*D
    float* idxo = zq + (size_t)BZ * TZ * DZ;                // B*N_CB*T
    float* lg   = idxo + (size_t)BZ * N_CB * TZ;            // B*N_CB*T*K
    float* cbn  = (float*)d_ws;                             // 8*1024 floats

    hipLaunchKernelGGL(rvq_cbnorm_kernel,
                       dim3((N_CB * KENT + 255) / 256), dim3(256), 0, stream,
                       cbs, cbn);
    hipLaunchKernelGGL(rvq_main_kernel,
                       dim3(TOK_TOTAL / TOK_PER_BLOCK), dim3(256), 0, stream,
                       z, cbs, cbn, zq, idxo, lg);
}